// RealLRURTRL_34969623724360
// MI455X (gfx1250) — compile-verified
//
#include <hip/hip_runtime.h>
#include <hip/hip_bf16.h>
#include <math.h>

// Problem sizes (fixed by the reference): S=256, BT=32, D=N=O=128.
#define S_LEN 256
#define BT_DIM 32
#define D_DIM 128
#define N_DIM 128
#define O_DIM 128
#define ROWS (S_LEN * BT_DIM)       // 8192 flattened (s,b) rows
#define EPS_V 1e-7f

typedef __attribute__((ext_vector_type(2))) float v2f;
typedef __attribute__((ext_vector_type(8))) float v8f;

// out[r, n] = scale(n) * sum_d in[r, d] * W[n, d]        (i.e. in @ W^T)
//   in : [ROWS, 128] f32 row-major
//   W  : [128, 128] f32 row-major (W[n*128 + d])
//   APPLY_GAMMA: scale column n by gamma(n) = sqrt(1 - lam^2 + eps),
//                lam = exp(-exp(lambda_log[n])).
//
// One wave computes a 16-row strip across all 128 output columns
// (8 tiles of 16x16), accumulating with V_WMMA_F32_16X16X4_F32 over K=128.
// A fragment (16x4 f32, ISA 7.12.2): lanes 0-15 hold M=0..15, lane-half h
// and VGPR j hold K = 2h + j. B fragment (4x16) mirrors that striping:
// VGPR j, lane l: N = l%16, K = j + 2*(l/16).
template <bool APPLY_GAMMA>
__global__ __launch_bounds__(256) void gemm_nt_wmma(
    const float* __restrict__ in, const float* __restrict__ W,
    const float* __restrict__ lambda_log, float* __restrict__ out) {
  const int lane = threadIdx.x & 31;
  const int wave = blockIdx.x * 8 + (threadIdx.x >> 5);
  const int rowBase = wave * 16;          // 512 waves cover 8192 rows
  const int half = lane >> 4;             // 0: lanes 0-15, 1: lanes 16-31
  const int lane16 = lane & 15;

  v8f acc[8];
  const v8f zero = {0.f, 0.f, 0.f, 0.f, 0.f, 0.f, 0.f, 0.f};
#pragma unroll
  for (int t = 0; t < 8; ++t) acc[t] = zero;

  const float* __restrict__ arow = in + (size_t)(rowBase + lane16) * D_DIM;

  for (int k0 = 0; k0 < D_DIM; k0 += 4) {
    const int ka = k0 + 2 * half;         // K index for vgpr 0 of the frags
    v2f a;
    a.x = arow[ka];
    a.y = arow[ka + 1];
#pragma unroll
    for (int t = 0; t < 8; ++t) {
      const int n = t * 16 + lane16;
      v2f b;
      b.x = W[(size_t)n * D_DIM + ka];
      b.y = W[(size_t)n * D_DIM + ka + 1];
      // 8 args: (neg_a, A, neg_b, B, c_mod, C, reuse_a, reuse_b)
      acc[t] = __builtin_amdgcn_wmma_f32_16x16x4_f32(
          false, a, false, b, (short)0, acc[t], false, false);
    }
  }

  // C/D layout: VGPR r, lanes 0-15 -> M=r, lanes 16-31 -> M=r+8; N = lane%16.
#pragma unroll
  for (int t = 0; t < 8; ++t) {
    const int n = t * 16 + lane16;
    float g = 1.0f;
    if (APPLY_GAMMA) {
      const float lam = __expf(-__expf(lambda_log[n]));
      g = sqrtf(fmaf(-lam, lam, 1.0f + EPS_V));
    }
#pragma unroll
    for (int r = 0; r < 8; ++r) {
      const int row = rowBase + r + 8 * half;
      out[(size_t)row * N_DIM + n] = g * acc[t][r];
    }
  }
}

// Diagonal linear recurrence, fully independent per (b, n) element:
//   h_s = lam[n] * h_{s-1} + U[s, b, n],   U overwritten in place with h_s.
// 4096 threads; the loads of U[s] do not depend on the FMA chain, so they
// pipeline ahead of the 256 dependent fmaf's (the true critical path).
__global__ __launch_bounds__(256) void scan_kernel(
    const float* __restrict__ lambda_log, float* __restrict__ U,
    float* __restrict__ hT_out) {
  const int i = blockIdx.x * blockDim.x + threadIdx.x;   // 0..4095 = b*128+n
  const int n = i & (N_DIM - 1);
  const float lam = __expf(-__expf(lambda_log[n]));
  float h = 0.0f;
  float* __restrict__ p = U + i;
#pragma unroll 8
  for (int s = 0; s < S_LEN; ++s) {
    const float u = p[(size_t)s * (BT_DIM * N_DIM)];
    h = fmaf(lam, h, u);
    p[(size_t)s * (BT_DIM * N_DIM)] = h;
  }
  hT_out[i] = h;   // hT is [BT, N] flat = i
}

extern "C" void kernel_launch(void* const* d_in, const int* in_sizes, int n_in,
                              void* d_out, int out_size, void* d_ws,
                              size_t ws_size, hipStream_t stream) {
  (void)in_sizes; (void)n_in; (void)out_size; (void)ws_size;
  const float* x          = (const float*)d_in[0];  // [256, 32, 128]
  const float* lambda_log = (const float*)d_in[1];  // [128]
  const float* B          = (const float*)d_in[2];  // [128, 128]
  const float* C          = (const float*)d_in[3];  // [128, 128]
  float* out = (float*)d_out;                       // outputs [256,32,128] ++ hT [32,128]
  float* U   = (float*)d_ws;                        // 8192*128 f32 = 4 MB scratch

  float* outputs = out;                       // 1,048,576 elements
  float* hT      = out + (size_t)ROWS * O_DIM;  // 4,096 elements

  // 1) U = gamma ⊙ (X · B^T)
  gemm_nt_wmma<true><<<dim3(ROWS / 16 / 8), dim3(256), 0, stream>>>(
      x, B, lambda_log, U);
  // 2) in-place diagonal scan over S; also emits hT
  scan_kernel<<<dim3((BT_DIM * N_DIM) / 256), dim3(256), 0, stream>>>(
      lambda_log, U, hT);
  // 3) outputs = hs · C^T
  gemm_nt_wmma<false><<<dim3(ROWS / 16 / 8), dim3(256), 0, stream>>>(
      U, C, lambda_log, outputs);
}